// KinematicModule_25082609008677
// MI455X (gfx1250) — compile-verified
//
#include <hip/hip_runtime.h>
#include <math.h>

#define D_LVL 32
#define M_ATM 32768
#define NATM (1 + D_LVL * M_ATM)

typedef float v2f __attribute__((ext_vector_type(2)));
typedef float v8f __attribute__((ext_vector_type(8)));

// R = Rz(ga) * Ry(be) * Rx(al), row-major 3x3. DOFs are in [0,1) rad -> hw sin/cos fine.
__device__ __forceinline__ void rzyx(float al, float be, float ga, float* R) {
    float sa = __sinf(al), ca = __cosf(al);
    float sb = __sinf(be), cb = __cosf(be);
    float sg = __sinf(ga), cg = __cosf(ga);
    R[0] = cg * cb; R[1] = cg * sb * sa - sg * ca; R[2] = cg * sb * ca + sg * sa;
    R[3] = sg * cb; R[4] = sg * sb * sa + cg * ca; R[5] = sg * sb * ca - cg * sa;
    R[6] = -sb;     R[7] = cb * sa;                R[8] = cb * ca;
}

// Local homogeneous transform, 3x4 row-major (bottom row [0,0,0,1] implicit).
// Bond (dtype!=0): rot_x(d0) @ rot_z(d1) @ trans_x(d2) @ rot_x(d3), expanded analytically.
// Jump (dtype==0): R = Rzyx(d3,d4,d5) @ Rzyx(d6,d7,d8), t = d0..d2.
__device__ __forceinline__ void local_ht(const float* __restrict__ dof, int dtype, float* H) {
    if (dtype != 0) {
        float a = dof[0], t = dof[1], d = dof[2], c = dof[3];
        float sa = __sinf(a), ca = __cosf(a);
        float st = __sinf(t), ct = __cosf(t);
        float sc = __sinf(c), cc = __cosf(c);
        H[0] = ct;      H[1] = -st * cc;               H[2]  = st * sc;                 H[3]  = d * ct;
        H[4] = ca * st; H[5] = ca * ct * cc - sa * sc; H[6]  = -ca * ct * sc - sa * cc; H[7]  = d * ca * st;
        H[8] = sa * st; H[9] = sa * ct * cc + ca * sc; H[10] = -sa * ct * sc + ca * cc; H[11] = d * sa * st;
    } else {
        float R1[9], R2[9];
        rzyx(dof[3], dof[4], dof[5], R1);
        rzyx(dof[6], dof[7], dof[8], R2);
#pragma unroll
        for (int r = 0; r < 3; ++r) {
#pragma unroll
            for (int cn = 0; cn < 3; ++cn) {
                H[r * 4 + cn] = R1[r * 3 + 0] * R2[0 * 3 + cn]
                              + R1[r * 3 + 1] * R2[1 * 3 + cn]
                              + R1[r * 3 + 2] * R2[2 * 3 + cn];
            }
            H[r * 4 + 3] = dof[r];
        }
    }
}

// One-thread kernel: root's global frame = its own (jump) local HT.
__global__ void root_kernel(const float* __restrict__ dofs,
                            const int* __restrict__ doftype,
                            float* __restrict__ Groot,
                            float* __restrict__ out) {
    if (threadIdx.x == 0 && blockIdx.x == 0) {
        float H[12];
        local_ht(dofs, doftype[0], H);
        for (int i = 0; i < 12; ++i) Groot[i] = H[i];
        out[0] = H[3]; out[1] = H[7]; out[2] = H[11];
    }
}

// One generation. Each wave handles 32 atoms:
//   Phase 1: lane i gathers parent frame + builds local HT of atom i -> LDS slabs.
//   Phase 2: 8x V_WMMA_F32_16X16X4_F32 (branch-free, EXEC all-1s):
//       A (16x4) = [P0;P1;P2;P3] stacked parent homogeneous HTs (ds_load_b64)
//       B (4x16) = [L0|L1|L2|L3] local homogeneous HTs           (ds_load_b32 x2)
//       D diagonal 4x4 blocks = P_i @ L_i -> staged to LDS (dump slot for idle lanes)
//   Phase 3: lane i writes atom i's composed frame (3x global_store_b128, coalesced)
//            and its coordinates (the .w column).
__global__ __launch_bounds__(256)
void level_kernel(const float* __restrict__ dofs,
                  const int* __restrict__ level_nodes,
                  const int* __restrict__ level_parents,
                  const int* __restrict__ doftype,
                  const float* __restrict__ Gprev,
                  float* __restrict__ Gcur,
                  const float* __restrict__ Groot,
                  float* __restrict__ out,
                  int lvl) {
    __shared__ float sP[8][32 * 12];   // parent frames, per wave
    __shared__ float sH[8][32 * 12];   // local HTs, per wave
    __shared__ float sG[8][32 * 12];   // composed frames, per wave
    __shared__ float sDump[8][48];     // write sink for non-diagonal lanes
    __shared__ int   sN[8][32];        // node ids, per wave

    const int lane = threadIdx.x & 31;
    const int wave = threadIdx.x >> 5;
    const int atomBase = blockIdx.x * 256 + wave * 32;   // this wave's first atom-in-level
    const int lbase = lvl * M_ATM;

    // ---------------- Phase 1: one atom per lane ----------------
    {
        const int aj   = atomBase + lane;
        const int node = level_nodes[lbase + aj];
        const int par  = level_parents[lbase + aj];
        sN[wave][lane] = node;

        const int prevBase = 1 + (lvl - 1) * M_ATM;      // parents live in previous generation
        const float4* Gp4 = (par == 0)
            ? (const float4*)Groot
            : (const float4*)(Gprev + (size_t)(par - prevBase) * 12);
        float4 p0 = Gp4[0], p1 = Gp4[1], p2 = Gp4[2];
        float4* sPl = (float4*)&sP[wave][lane * 12];
        sPl[0] = p0; sPl[1] = p1; sPl[2] = p2;

        alignas(16) float H[12];
        local_ht(dofs + (size_t)node * 9, doftype[node], H);
        float4* sHl = (float4*)&sH[wave][lane * 12];
        sHl[0] = ((float4*)H)[0];
        sHl[1] = ((float4*)H)[1];
        sHl[2] = ((float4*)H)[2];
    }

    __syncthreads();   // cross-lane LDS handoff

    // ---------------- Phase 2: 8 quads of 4 atoms, one WMMA each ----------------
    const int half = lane >> 4;        // 0: K/rows {0,1}; 1: K/rows {2,3}
    const int lm   = lane & 15;        // A row index == B col index == D col index
    const int blk  = lm >> 2;          // which atom of the quad this lane serves
    const int sub  = lm & 3;           // row-in-block (A) / col-in-block (B/D)

    // Loop-invariant operand plumbing (all blends, no control flow in the loop):
    const bool  subLT3  = (sub < 3);
    const float aConstY = (half == 1) ? 1.0f : 0.0f;            // bottom row, k in {2,3}
    const float bConstY = (half == 1 && sub == 3) ? 1.0f : 0.0f; // B row 3 element
    const bool  bLoY    = (half == 0);
    const bool  oddBlk  = (blk & 1) != 0;
    const bool  useful  = (half == (blk >> 1));                  // lane holds diagonal data
    const int   aOff  = (subLT3 ? sub : 2) * 4 + half * 2;       // clamped, 8B-aligned pair
    const int   bxOff = half * 8 + sub;
    const int   byOff = 4 + sub;

    const float* Pw = &sP[wave][0];
    const float* Hw = &sH[wave][0];
    float* Gw = &sG[wave][0];
    float* dumpBase = &sDump[wave][lane];                        // distinct dword per lane

#pragma unroll
    for (int q = 0; q < 8; ++q) {
        const int la12 = (q * 4 + blk) * 12;                     // wave-local atom * 12

        v2f av = *(const v2f*)&Pw[la12 + aOff];                  // ds_load_b64
        v2f a;
        a.x = subLT3 ? av.x : 0.0f;
        a.y = subLT3 ? av.y : aConstY;

        v2f b;
        b.x = Hw[la12 + bxOff];
        float byv = Hw[la12 + byOff];
        b.y = bLoY ? byv : bConstY;

        v8f c = {};
        // 8 args: (neg_a, A, neg_b, B, c_mod, C, reuse_a, reuse_b)
        v8f d = __builtin_amdgcn_wmma_f32_16x16x4_f32(false, a, false, b, (short)0, c, false, false);

        // D layout: VGPR v, lanes 0-15 -> row v, lanes 16-31 -> row v+8; col = lm.
        // Diagonal block rows 4*blk..4*blk+2 live in VGPRs {0,1,2} (even blk) or {4,5,6} (odd).
        float r0 = oddBlk ? d[4] : d[0];
        float r1 = oddBlk ? d[5] : d[1];
        float r2 = oddBlk ? d[6] : d[2];
        float* g = useful ? (Gw + la12 + sub) : dumpBase;        // pointer blend, no branch
        g[0] = r0; g[4] = r1; g[8] = r2;
    }

    __syncthreads();   // composed frames now complete in sG

    // ---------------- Phase 3: coalesced writeback, one atom per lane ----------------
    {
        const float4* sGl = (const float4*)&sG[wave][lane * 12];
        float4 g0 = sGl[0], g1 = sGl[1], g2 = sGl[2];
        float4* Gout = (float4*)(Gcur + (size_t)(atomBase + lane) * 12);
        Gout[0] = g0; Gout[1] = g1; Gout[2] = g2;                // 3x global_store_b128

        const int nd = sN[wave][lane];
        float* o = out + (size_t)nd * 3;
        o[0] = g0.w; o[1] = g1.w; o[2] = g2.w;                   // translation column
    }
}

extern "C" void kernel_launch(void* const* d_in, const int* in_sizes, int n_in,
                              void* d_out, int out_size, void* d_ws, size_t ws_size,
                              hipStream_t stream) {
    const float* dofs        = (const float*)d_in[0];
    const int* level_nodes   = (const int*)d_in[1];
    const int* level_parents = (const int*)d_in[2];
    const int* doftype       = (const int*)d_in[3];
    float* out = (float*)d_out;

    // Workspace: root frame (16 floats) + two ping-pong generation frame buffers
    // (M*12 floats each, ~1.57 MB -> both stay L2-resident across the 32 levels).
    float* Groot = (float*)d_ws;
    float* Gbuf0 = Groot + 16;
    float* Gbuf1 = Gbuf0 + (size_t)M_ATM * 12;

    root_kernel<<<1, 32, 0, stream>>>(dofs, doftype, Groot, out);

    for (int l = 0; l < D_LVL; ++l) {
        float* Gcur        = (l & 1) ? Gbuf1 : Gbuf0;
        const float* Gprev = (l & 1) ? Gbuf0 : Gbuf1;   // unused (all parents==root) when l==0
        level_kernel<<<M_ATM / 256, 256, 0, stream>>>(
            dofs, level_nodes, level_parents, doftype,
            Gprev, Gcur, Groot, out, l);
    }
}